// SSIMLoss_3478923510398
// MI455X (gfx1250) — compile-verified
//
#include <hip/hip_runtime.h>

// SSIM loss for 16x3x512x512 fp32 images on gfx1250 (MI455X).
// Separable 11x11 Gaussian conv as banded matmuls on V_WMMA_F32_16X16X4_F32:
//   Out16x16 = Gv(16x28) x Patch(28x28pad) x Gh(28x16)
// v2: stage-1 fused over the 5 SSIM quantities (patch read once per K-chunk),
//     transposed patch + row-major T in LDS so every WMMA operand is a single
//     aligned ds_load_b64 into an even VGPR pair (no v_mov repacking),
//     fast reciprocal + 1 NR step instead of IEEE divide.

typedef float v2f __attribute__((ext_vector_type(2)));
typedef float v8f __attribute__((ext_vector_type(8)));

#define IMG_H 512
#define IMG_W 512
#define TILE  16
#define PROWS 28   // 26 patch rows (16 + 2*5 halo) padded to 28 (K mult of 4)
#define PCT   32   // transposed-patch columns (26 valid, rest zero)
#define KCH   7    // 28 / 4 K-chunks
#define WPB   4    // waves per block
#define SSIM_C1 0.0001f
#define SSIM_C2 0.0009f

// gaussian(sigma=1.5, win=11), normalized
__device__ __constant__ float GW[11] = {
    1.02838e-3f, 7.59866e-3f, 3.600077e-2f, 1.0936069e-1f, 2.1300528e-1f,
    2.6601172e-1f, 2.1300528e-1f, 1.0936069e-1f, 3.600077e-2f, 7.59866e-3f,
    1.02838e-3f};

__device__ __forceinline__ float gwt(int d) {
    return (d >= 0 && d < 11) ? GW[d] : 0.0f;
}

__device__ __forceinline__ float fast_recip(float d) {
    float r = __builtin_amdgcn_rcpf(d);
    float e = fmaf(-d, r, 1.0f);      // one Newton-Raphson step
    return fmaf(r, e, r);
}

__global__ void __launch_bounds__(WPB * 32)
ssim_tiles(const float* __restrict__ img1, const float* __restrict__ img2,
           float* __restrict__ accum) {
    // transposed patches: [col][row] so B-operand K pairs are contiguous b64
    __shared__ float sP1[WPB][PCT][PROWS];
    __shared__ float sP2[WPB][PCT][PROWS];
    // five T planes, row-major [m][kcol] so A-operand K pairs are contiguous
    __shared__ float sT[WPB][5][TILE][PROWS];
    __shared__ float sRed[WPB];

    const int wave = threadIdx.x >> 5;
    const int lane = threadIdx.x & 31;
    const int m    = lane & 15;  // M for A / N for B / N for D
    const int hi   = lane >> 4;  // lane half: K pair select (A/B), M+8 (D)

    // one 16x16 output tile per wave; 32x32 tiles per 512x512 plane
    const int tile_id = blockIdx.x * WPB + wave;
    const int plane   = tile_id >> 10;
    const int t       = tile_id & 1023;
    const int ty      = t >> 5;
    const int tx      = t & 31;
    const int base    = plane * (IMG_H * IMG_W);
    const int y0      = ty * TILE - 5;
    const int x0      = tx * TILE - 5;

    // coalesced global read (lane = column), transposed LDS write
#pragma unroll
    for (int r = 0; r < PROWS; ++r) {
        const int gy  = y0 + r;
        const int gx  = x0 + lane;
        const bool ok = (r < 26) && (lane < 26) && (gy >= 0) && (gy < IMG_H) &&
                        (gx >= 0) && (gx < IMG_W);
        const int idx = base + gy * IMG_W + gx;
        sP1[wave][lane][r] = ok ? img1[idx] : 0.0f;
        sP2[wave][lane][r] = ok ? img2[idx] : 0.0f;
    }
    __syncthreads();

    // Banded Gaussian operand: identical per-lane indexing as 32-bit WMMA
    // A matrix (stage-1 Gv) and B matrix (stage-2 Gh) -> one register set.
    v2f G[KCH];
#pragma unroll
    for (int c = 0; c < KCH; ++c) {
        const int kk = 4 * c + 2 * hi;
        G[c][0] = gwt(kk - m);
        G[c][1] = gwt(kk + 1 - m);
    }

    // ---- stage 1 (vertical pass), fused across the 5 quantities ----
#pragma unroll
    for (int n = 0; n < 2; ++n) {
        v8f acc[5];
#pragma unroll
        for (int q = 0; q < 5; ++q)
            acc[q] = (v8f){0.f, 0.f, 0.f, 0.f, 0.f, 0.f, 0.f, 0.f};

        const int col = 16 * n + m;  // patch column this lane feeds (B's N)
#pragma unroll
        for (int c = 0; c < KCH; ++c) {
            const int kk = 4 * c + 2 * hi;
            const v2f b1 = *(const v2f*)&sP1[wave][col][kk];  // ds_load_b64
            const v2f b2 = *(const v2f*)&sP2[wave][col][kk];  // ds_load_b64
            acc[0] = __builtin_amdgcn_wmma_f32_16x16x4_f32(
                false, G[c], false, b1, (short)0, acc[0], false, false);
            acc[1] = __builtin_amdgcn_wmma_f32_16x16x4_f32(
                false, G[c], false, b2, (short)0, acc[1], false, false);
            acc[2] = __builtin_amdgcn_wmma_f32_16x16x4_f32(
                false, G[c], false, b1 * b1, (short)0, acc[2], false, false);
            acc[3] = __builtin_amdgcn_wmma_f32_16x16x4_f32(
                false, G[c], false, b2 * b2, (short)0, acc[3], false, false);
            acc[4] = __builtin_amdgcn_wmma_f32_16x16x4_f32(
                false, G[c], false, b1 * b2, (short)0, acc[4], false, false);
        }
        // D layout: vgpr r -> row (r + 8*hi), col (16n + m). Cols >= 28 are
        // all-zero (patch pad) and dead weight in stage 2 -> skip the store.
        if (col < PROWS) {
#pragma unroll
            for (int q = 0; q < 5; ++q)
#pragma unroll
                for (int r = 0; r < 8; ++r)
                    sT[wave][q][r + 8 * hi][col] = acc[q][r];
        }
    }
    __syncthreads();

    // ---- stage 2 (horizontal pass): Out_q = T_q x Gh ----
    v8f res[5];
#pragma unroll
    for (int q = 0; q < 5; ++q) {
        v8f out = {0.f, 0.f, 0.f, 0.f, 0.f, 0.f, 0.f, 0.f};
#pragma unroll
        for (int c = 0; c < KCH; ++c) {
            const v2f a = *(const v2f*)&sT[wave][q][m][4 * c + 2 * hi];
            out = __builtin_amdgcn_wmma_f32_16x16x4_f32(
                false, a, false, G[c], (short)0, out, false, false);
        }
        res[q] = out;
    }

    // ---- elementwise SSIM on the 8 D-values this lane owns ----
    float sum = 0.0f;
#pragma unroll
    for (int r = 0; r < 8; ++r) {
        const float mu1 = res[0][r], mu2 = res[1][r];
        const float e11 = res[2][r], e22 = res[3][r], e12 = res[4][r];
        const float mu1s = mu1 * mu1, mu2s = mu2 * mu2, mu12 = mu1 * mu2;
        const float s11 = e11 - mu1s, s22 = e22 - mu2s, s12 = e12 - mu12;
        const float num = (2.0f * mu12 + SSIM_C1) * (2.0f * s12 + SSIM_C2);
        const float den = (mu1s + mu2s + SSIM_C1) * (s11 + s22 + SSIM_C2);
        sum = fmaf(num, fast_recip(den), sum);
    }

    // wave32 butterfly reduce, then one atomic per block
#pragma unroll
    for (int off = 16; off >= 1; off >>= 1)
        sum += __shfl_xor(sum, off, 32);
    if (lane == 0) sRed[wave] = sum;
    __syncthreads();
    if (threadIdx.x == 0) {
        float bs = 0.0f;
#pragma unroll
        for (int w = 0; w < WPB; ++w) bs += sRed[w];
        atomicAdd(accum, bs);
    }
}

__global__ void ssim_zero(float* a) { a[0] = 0.0f; }

__global__ void ssim_finalize(const float* a, float* out, float inv_n) {
    out[0] = 1.0f - a[0] * inv_n;
}

extern "C" void kernel_launch(void* const* d_in, const int* in_sizes, int n_in,
                              void* d_out, int out_size, void* d_ws,
                              size_t ws_size, hipStream_t stream) {
    (void)n_in; (void)out_size; (void)ws_size;
    const float* img1 = (const float*)d_in[0];
    const float* img2 = (const float*)d_in[1];
    float* out = (float*)d_out;
    float* acc = (float*)d_ws;

    const int n      = in_sizes[0];  // 16*3*512*512
    const int planes = n / (IMG_H * IMG_W);
    const int tiles  = planes * ((IMG_H / TILE) * (IMG_W / TILE));
    const int blocks = tiles / WPB;

    ssim_zero<<<1, 1, 0, stream>>>(acc);
    ssim_tiles<<<blocks, WPB * 32, 0, stream>>>(img1, img2, acc);
    ssim_finalize<<<1, 1, 0, stream>>>(acc, out, 1.0f / (float)n);
}